// AttentiveStudentModel_73933567034011
// MI455X (gfx1250) — compile-verified
//
#include <hip/hip_runtime.h>
#include <math.h>

// Problem constants from the reference
#define LEN      200       // MAX_LEN
#define LENP     208       // padded to 13 tiles of 16
#define DIM      64
#define NHEAD    2
#define INV_T    10.0f     // 1 / TEMPERATURE
#define ESTRIDE  68        // LDS row stride in floats (bank-conflict padding, 16B aligned)

typedef __attribute__((ext_vector_type(2))) float v2f;
typedef __attribute__((ext_vector_type(8))) float v8f;

__global__ __launch_bounds__(256) void attn_student_kernel(
    const int* __restrict__ hist,      // [B, LEN]
    const float* __restrict__ table,   // [NUM_ITEMS, DIM]
    const float* __restrict__ queries, // [NHEAD, DIM]
    float* __restrict__ out)           // [B, DIM]
{
    __shared__ __align__(16) float s_emb[LENP * ESTRIDE]; // gathered rows (zeros for pad)
    __shared__ float s_q[NHEAD * DIM];
    __shared__ float s_score[NHEAD * LENP];               // RAW scores (mask applied later)
    __shared__ float s_p[NHEAD * LENP];                   // unnormalized softmax weights
    __shared__ float s_inv[NHEAD];                        // 1/Z per head
    __shared__ int   s_idx[LENP];

    const int b   = blockIdx.x;
    const int tid = threadIdx.x;
    const int wv  = tid >> 5;   // wave id 0..7
    const int ln  = tid & 31;   // lane id
    const int hl  = ln >> 4;    // half-wave: 0 = lanes 0-15, 1 = lanes 16-31
    const int lml = ln & 15;    // lane mod 16

    // ---- Phase A: stage indices + queries ----------------------------------
    if (tid < NHEAD * DIM) s_q[tid] = queries[tid];
    for (int l = tid; l < LENP; l += 256)
        s_idx[l] = (l < LEN) ? hist[(size_t)b * LEN + l] : -1;
    __syncthreads();

    // ---- Phase B: gather rows into LDS (skip HBM traffic for padding) ------
    for (int i = tid; i < LENP * (DIM / 4); i += 256) {
        int row = i >> 4, seg = i & 15;
        int id = s_idx[row];
        float4 val = make_float4(0.f, 0.f, 0.f, 0.f);
        if (id >= 0)
            val = *(const float4*)(table + (size_t)id * DIM + seg * 4);
        *(float4*)(&s_emb[row * ESTRIDE + seg * 4]) = val;
    }
    __syncthreads();

    // ---- Phase C: scores[l,k] = emb[l,:] . q[k,:] via v_wmma_f32_16x16x4_f32
    // B fragment (query side) is identical for every M-tile: hoist into regs.
    // B layout (4x16): vgpr v holds row k_inner = hl*2+v, col = lml.
    v2f bq[DIM / 4];
    for (int c = 0; c < DIM / 4; ++c) {
        v2f tmp;
        for (int v = 0; v < 2; ++v) {
            int dd = c * 4 + hl * 2 + v;
            tmp[v] = (lml < NHEAD) ? s_q[lml * DIM + dd] : 0.0f;
        }
        bq[c] = tmp;
    }
    for (int mt = wv; mt < LENP / 16; mt += 8) {   // 13 M-tiles over 8 waves
        v8f acc = {};
        for (int c = 0; c < DIM / 4; ++c) {
            // A layout (16x4): lane row m = lml, vgpr v holds k_inner = hl*2+v
            v2f a;
            int row = mt * 16 + lml;
            int dd  = c * 4 + hl * 2;
            a[0] = s_emb[row * ESTRIDE + dd];
            a[1] = s_emb[row * ESTRIDE + dd + 1];
            acc = __builtin_amdgcn_wmma_f32_16x16x4_f32(
                false, a, false, bq[c], (short)0, acc, false, false);
        }
        // C layout: vgpr r -> row M = r + hl*8, col = lml. Columns 0..1 = heads.
        // Store RAW scores; validity mask is applied in Phase D (keeps this
        // path branch-free: pad rows of s_emb are zero so acc is finite).
        if (lml < NHEAD) {
            int base = lml * LENP + mt * 16 + hl * 8;
            for (int r = 0; r < 8; ++r)
                s_score[base + r] = acc[r];
        }
    }
    __syncthreads();

    // ---- Phase D: mask + softmax numerators + 1/Z (one wave per head) ------
    if (wv < NHEAD) {
        int k = wv;
        float m = -1.0e30f;
        for (int l = ln; l < LENP; l += 32) {
            bool valid = (l < LEN) && (s_idx[l] >= 0);
            float sc = valid ? s_score[k * LENP + l] : -1.0e9f;
            m = fmaxf(m, sc);
        }
        for (int off = 16; off; off >>= 1) m = fmaxf(m, __shfl_xor(m, off, 32));
        float z = 0.f;
        for (int l = ln; l < LENP; l += 32) {
            bool valid = (l < LEN) && (s_idx[l] >= 0);
            float sc = valid ? s_score[k * LENP + l] : -1.0e9f;
            float p = (l < LEN) ? __expf((sc - m) * INV_T) : 0.0f;
            s_p[k * LENP + l] = p;
            z += p;
        }
        for (int off = 16; off; off >>= 1) z += __shfl_xor(z, off, 32);
        if (ln == 0) s_inv[k] = 1.0f / z;
    }
    __syncthreads();

    // ---- Phase E: head[k,d] = sum_l p[k,l]*emb[l,d] via WMMA ---------------
    // 4 N-tiles of 16 over d; waves 0..3; 50 K-chunks of 4 over l.
    if (wv < DIM / 16) {
        v8f acc = {};
        for (int c = 0; c < LEN / 4; ++c) {
            v2f a, bb;
            for (int v = 0; v < 2; ++v) {
                int l = c * 4 + hl * 2 + v;           // k_inner = hl*2+v
                a[v]  = (lml < NHEAD) ? s_p[lml * LENP + l] : 0.0f; // A row = head
                bb[v] = s_emb[l * ESTRIDE + wv * 16 + lml];         // B col = d
            }
            acc = __builtin_amdgcn_wmma_f32_16x16x4_f32(
                false, a, false, bb, (short)0, acc, false, false);
        }
        // Rows 0,1 of C live in acc[0],acc[1] on lanes 0-15 (hl==0).
        if (hl == 0) {
            int d = wv * 16 + lml;
            out[(size_t)b * DIM + d] = acc[0] * s_inv[0] + acc[1] * s_inv[1];
        }
    }
}

extern "C" void kernel_launch(void* const* d_in, const int* in_sizes, int n_in,
                              void* d_out, int out_size, void* d_ws, size_t ws_size,
                              hipStream_t stream) {
    const int*   hist    = (const int*)d_in[0];
    const float* table   = (const float*)d_in[1];
    const float* queries = (const float*)d_in[2];
    float*       out     = (float*)d_out;

    const int B = in_sizes[0] / LEN;   // 4096
    hipLaunchKernelGGL(attn_student_kernel, dim3(B), dim3(256), 0, stream,
                       hist, table, queries, out);
}